// GNNEncoder_21930103013914
// MI455X (gfx1250) — compile-verified
//
#include <hip/hip_runtime.h>

typedef __attribute__((ext_vector_type(2))) float v2f;
typedef __attribute__((ext_vector_type(8))) float v8f;

#define N_NODES 100000
#define N_EDGES 1600000
#define D 64
#define N_LAYERS 5

static inline size_t align256(size_t x) { return (x + 255) & ~(size_t)255; }

// ---------------- degree (int) ----------------
__global__ __launch_bounds__(256) void gnn_deg_int_kernel(const int* __restrict__ dst,
                                                          int* __restrict__ degi, int nE) {
  int e = blockIdx.x * 256 + threadIdx.x;
  if (e < nE) atomicAdd(&degi[dst[e]], 1);
}

__global__ __launch_bounds__(256) void gnn_invdeg_from_int_kernel(const int* __restrict__ degi,
                                                                  float* __restrict__ invd, int n) {
  int i = blockIdx.x * 256 + threadIdx.x;
  if (i < n) invd[i] = 1.0f / fmaxf((float)degi[i], 1.0f);
}

// ---------------- prefix scan for CSR rowptr (3 phases, graph-safe) ----------------
__global__ __launch_bounds__(256) void gnn_scan_p1_kernel(const int* __restrict__ degi,
                                                          int* __restrict__ blockSums, int n) {
  __shared__ int s[256];
  int i = blockIdx.x * 256 + threadIdx.x;
  s[threadIdx.x] = (i < n) ? degi[i] : 0;
  __syncthreads();
  for (int off = 128; off > 0; off >>= 1) {
    if (threadIdx.x < off) s[threadIdx.x] += s[threadIdx.x + off];
    __syncthreads();
  }
  if (threadIdx.x == 0) blockSums[blockIdx.x] = s[0];
}

__global__ void gnn_scan_p2_kernel(int* __restrict__ blockSums, int nb) {
  if (blockIdx.x == 0 && threadIdx.x == 0) {
    int acc = 0;
    for (int i = 0; i < nb; ++i) {
      int v = blockSums[i];
      blockSums[i] = acc;
      acc += v;
    }
  }
}

__global__ __launch_bounds__(256) void gnn_scan_p3_kernel(const int* __restrict__ degi,
                                                          const int* __restrict__ blockSums,
                                                          int* __restrict__ rowptr,
                                                          int* __restrict__ fillpos, int n) {
  __shared__ int s[256];
  int i = blockIdx.x * 256 + threadIdx.x;
  int v = (i < n) ? degi[i] : 0;
  s[threadIdx.x] = v;
  __syncthreads();
  for (int off = 1; off < 256; off <<= 1) {
    int t = (threadIdx.x >= off) ? s[threadIdx.x - off] : 0;
    __syncthreads();
    s[threadIdx.x] += t;
    __syncthreads();
  }
  if (i < n) {
    int excl = s[threadIdx.x] - v + blockSums[blockIdx.x];
    rowptr[i] = excl;
    fillpos[i] = excl;
  }
}

__global__ __launch_bounds__(256) void gnn_csr_fill_kernel(const int* __restrict__ src,
                                                           const int* __restrict__ dstv,
                                                           int* __restrict__ fillpos,
                                                           int* __restrict__ colidx, int nE) {
  int e = blockIdx.x * 256 + threadIdx.x;
  if (e < nE) {
    int pos = atomicAdd(&fillpos[dstv[e]], 1);
    colidx[pos] = src[e];
  }
}

// ---------------- CSR gather aggregation: zero float atomics ----------------
// One wave per node; 32 lanes x float2 = the full 64-f row (256B coalesced / neighbor).
// Unroll-by-4 with independent accumulators => 4 L2 gathers in flight per wave.
__global__ __launch_bounds__(256) void gnn_gather_agg_kernel(const float* __restrict__ h,
                                                             const int* __restrict__ colidx,
                                                             const int* __restrict__ rowptr,
                                                             const int* __restrict__ degi,
                                                             float* __restrict__ agg, int nN) {
  int wave = threadIdx.x >> 5;
  int lane = threadIdx.x & 31;
  int node = blockIdx.x * 8 + wave;
  if (node >= nN) return;
  int beg = rowptr[node];
  int cnt = degi[node];
  const float* hp = h + 2 * lane;

  float2 a0 = make_float2(0.f, 0.f), a1 = a0, a2 = a0, a3 = a0;
  int t = 0;
  for (; t + 4 <= cnt; t += 4) {
    int n0 = colidx[beg + t + 0];
    int n1 = colidx[beg + t + 1];
    int n2 = colidx[beg + t + 2];
    int n3 = colidx[beg + t + 3];
    const float2 v0 = *(const float2*)(hp + (long)n0 * D);
    const float2 v1 = *(const float2*)(hp + (long)n1 * D);
    const float2 v2 = *(const float2*)(hp + (long)n2 * D);
    const float2 v3 = *(const float2*)(hp + (long)n3 * D);
    a0.x += v0.x; a0.y += v0.y;
    a1.x += v1.x; a1.y += v1.y;
    a2.x += v2.x; a2.y += v2.y;
    a3.x += v3.x; a3.y += v3.y;
  }
  for (; t < cnt; ++t) {
    int nb = colidx[beg + t];
    const float2 v = *(const float2*)(hp + (long)nb * D);
    a0.x += v.x; a0.y += v.y;
  }
  float2 acc;
  acc.x = (a0.x + a1.x) + (a2.x + a3.x);
  acc.y = (a0.y + a1.y) + (a2.y + a3.y);
  *(float2*)(agg + (long)node * D + 2 * lane) = acc;
}

// ---------------- fallback: edge-parallel atomic scatter ----------------
__global__ __launch_bounds__(256) void gnn_scatter_kernel(const float* __restrict__ h,
                                                          const int* __restrict__ src,
                                                          const int* __restrict__ dstv,
                                                          float* __restrict__ agg, int nE) {
  long tid = (long)blockIdx.x * 256 + threadIdx.x;
  int e = (int)(tid >> 4);
  int c = (int)(tid & 15);
  if (e >= nE) return;
  int s = src[e];
  int d = dstv[e];
  const float4 v = *(const float4*)(h + (long)s * D + c * 4);
  float* ap = agg + (long)d * D + c * 4;
  atomicAdd(ap + 0, v.x);
  atomicAdd(ap + 1, v.y);
  atomicAdd(ap + 2, v.z);
  atomicAdd(ap + 3, v.w);
}

// ---------------- fused SAGE layer GEMM via V_WMMA_F32_16X16X4_F32 ----------------
// out = relu?( (agg*invdeg)@Wl + b + h@Wr [+ h] )
// Block = 256 threads = 8 waves; each wave owns 16 output rows (4 tiles of 16x16).
// Weights staged in LDS pre-swizzled into WMMA B-fragment order:
// sWf[pass][k][tile][lane] = {B[kb][col], B[kb+1][col]}  -> one ds_load_b64 each.
__global__ __launch_bounds__(256) void gnn_sage_gemm_kernel(
    const float* __restrict__ agg, const float* __restrict__ h,
    const float* __restrict__ invdeg,
    const float* __restrict__ Wl, const float* __restrict__ Wr,
    const float* __restrict__ bias,
    float* __restrict__ out, int nN, int do_relu, int do_resid) {
  __shared__ __align__(16) float2 sWf[2][16][4][32];
  __shared__ __align__(16) float sB[D];
  int tid = threadIdx.x;

  // Stage + swizzle weights into fragment order (once per block, from L2).
  for (int t = tid; t < 2048; t += 256) {
    int k  = t >> 7;         // 0..15
    int n  = (t >> 5) & 3;   // 0..3
    int ln = t & 31;         // 0..31
    int koff = (ln < 16) ? 0 : 2;
    int lr = ln & 15;
    int kb = k * 4 + koff;
    int col = n * 16 + lr;
    float2 f0, f1;
    f0.x = Wl[(kb + 0) * D + col];
    f0.y = Wl[(kb + 1) * D + col];
    f1.x = Wr[(kb + 0) * D + col];
    f1.y = Wr[(kb + 1) * D + col];
    sWf[0][k][n][ln] = f0;
    sWf[1][k][n][ln] = f1;
  }
  if (tid < D) sB[tid] = bias[tid];
  __syncthreads();

  int wave = tid >> 5;
  int lane = tid & 31;
  int lrow = lane & 15;
  int khalf = (lane < 16) ? 0 : 2;  // ISA A-layout: lanes 0-15 -> K=4k,4k+1; 16-31 -> 4k+2,4k+3
  int rowbase = blockIdx.x * 128 + wave * 16;
  int M = rowbase + lrow;
  int Mc = (M < nN) ? M : (nN - 1);  // clamp: WMMA requires EXEC all-ones; stores guarded

  float invd = invdeg[Mc];

  v8f acc[4];
#pragma unroll
  for (int n = 0; n < 4; ++n) {
    float bv = sB[n * 16 + lrow];
#pragma unroll
    for (int j = 0; j < 8; ++j) acc[n][j] = bv;
  }

  for (int p = 0; p < 2; ++p) {
    const float* Arow = (p == 0 ? agg : h) + (long)Mc * D;
    float scale = (p == 0) ? invd : 1.0f;
#pragma unroll
    for (int k = 0; k < 16; ++k) {
      int kb = k * 4 + khalf;
      v2f a;
      a[0] = Arow[kb + 0] * scale;
      a[1] = Arow[kb + 1] * scale;
#pragma unroll
      for (int n = 0; n < 4; ++n) {
        float2 bf2 = sWf[p][k][n][lane];
        v2f bf;
        bf[0] = bf2.x;
        bf[1] = bf2.y;
        acc[n] = __builtin_amdgcn_wmma_f32_16x16x4_f32(
            false, a, false, bf, (short)0, acc[n], false, false);
      }
    }
  }

#pragma unroll
  for (int n = 0; n < 4; ++n) {
#pragma unroll
    for (int j = 0; j < 8; ++j) {
      int row = rowbase + ((lane < 16) ? j : j + 8);
      int col = n * 16 + lrow;
      if (row < nN) {
        float v = acc[n][j];
        if (do_resid) v += h[(long)row * D + col];
        if (do_relu) v = fmaxf(v, 0.0f);
        out[(long)row * D + col] = v;
      }
    }
  }
}

extern "C" void kernel_launch(void* const* d_in, const int* in_sizes, int n_in,
                              void* d_out, int out_size, void* d_ws, size_t ws_size,
                              hipStream_t stream) {
  const float* x  = (const float*)d_in[0];
  const int*   ei = (const int*)d_in[1];
  const float* Wl = (const float*)d_in[2];
  const float* Wr = (const float*)d_in[3];
  const float* b  = (const float*)d_in[4];
  float* out = (float*)d_out;

  const int* src = ei;            // edge_index[0]
  const int* dst = ei + N_EDGES;  // edge_index[1]

  const size_t featBytes = (size_t)N_NODES * D * sizeof(float);
  const size_t nodeI     = align256((size_t)N_NODES * sizeof(int));
  const size_t nodeF     = align256((size_t)N_NODES * sizeof(float));
  const size_t edgeI     = align256((size_t)N_EDGES * sizeof(int));
  const int scanBlocks   = (N_NODES + 255) / 256;
  const size_t bsumBytes = align256((size_t)scanBlocks * sizeof(int));

  const size_t csrNeed = 3 * nodeI + bsumBytes + nodeF + edgeI + 3 * featBytes;

  char* ws = (char*)d_ws;

  if (ws_size >= csrNeed) {
    // -------- CSR gather path: no float atomics in the per-layer loop --------
    int*   degi      = (int*)ws;   ws += nodeI;
    int*   rowptr    = (int*)ws;   ws += nodeI;
    int*   fillpos   = (int*)ws;   ws += nodeI;
    int*   blockSums = (int*)ws;   ws += bsumBytes;
    float* invdeg    = (float*)ws; ws += nodeF;
    int*   colidx    = (int*)ws;   ws += edgeI;
    float* agg       = (float*)ws; ws += featBytes;
    float* h1        = (float*)ws; ws += featBytes;
    float* h2        = (float*)ws;

    hipMemsetAsync(degi, 0, (size_t)N_NODES * sizeof(int), stream);
    gnn_deg_int_kernel<<<(N_EDGES + 255) / 256, 256, 0, stream>>>(dst, degi, N_EDGES);
    gnn_invdeg_from_int_kernel<<<scanBlocks, 256, 0, stream>>>(degi, invdeg, N_NODES);
    gnn_scan_p1_kernel<<<scanBlocks, 256, 0, stream>>>(degi, blockSums, N_NODES);
    gnn_scan_p2_kernel<<<1, 64, 0, stream>>>(blockSums, scanBlocks);
    gnn_scan_p3_kernel<<<scanBlocks, 256, 0, stream>>>(degi, blockSums, rowptr, fillpos, N_NODES);
    gnn_csr_fill_kernel<<<(N_EDGES + 255) / 256, 256, 0, stream>>>(src, dst, fillpos, colidx, N_EDGES);

    const float* h = x;
    float* houts[N_LAYERS] = {h1, h2, h1, h2, out};
    int gatherBlocks = (N_NODES + 7) / 8;
    int gemmBlocks   = (N_NODES + 127) / 128;

    for (int i = 0; i < N_LAYERS; ++i) {
      gnn_gather_agg_kernel<<<gatherBlocks, 256, 0, stream>>>(h, colidx, rowptr, degi, agg, N_NODES);
      int relu  = (i < N_LAYERS - 1) ? 1 : 0;
      int resid = (i > 0 && i < N_LAYERS - 1) ? 1 : 0;
      gnn_sage_gemm_kernel<<<gemmBlocks, 256, 0, stream>>>(
          agg, h, invdeg,
          Wl + (long)i * D * D, Wr + (long)i * D * D, b + (long)i * D,
          houts[i], N_NODES, relu, resid);
      h = houts[i];
    }
  } else {
    // -------- fallback: atomic scatter path (smaller workspace) --------
    float* invdeg = (float*)ws; ws += nodeF;
    float* agg    = (float*)ws; ws += featBytes;
    float* h1     = (float*)ws; ws += featBytes;
    float* h2     = (float*)ws;

    hipMemsetAsync(invdeg, 0, (size_t)N_NODES * sizeof(float), stream);
    gnn_deg_int_kernel<<<(N_EDGES + 255) / 256, 256, 0, stream>>>(dst, (int*)invdeg, N_EDGES);
    gnn_invdeg_from_int_kernel<<<scanBlocks, 256, 0, stream>>>((const int*)invdeg, invdeg, N_NODES);

    const float* h = x;
    float* houts[N_LAYERS] = {h1, h2, h1, h2, out};
    long scatterBlocks = ((long)N_EDGES * 16 + 255) / 256;
    int gemmBlocks = (N_NODES + 127) / 128;

    for (int i = 0; i < N_LAYERS; ++i) {
      hipMemsetAsync(agg, 0, featBytes, stream);
      gnn_scatter_kernel<<<(int)scatterBlocks, 256, 0, stream>>>(h, src, dst, agg, N_EDGES);
      int relu  = (i < N_LAYERS - 1) ? 1 : 0;
      int resid = (i > 0 && i < N_LAYERS - 1) ? 1 : 0;
      gnn_sage_gemm_kernel<<<gemmBlocks, 256, 0, stream>>>(
          agg, h, invdeg,
          Wl + (long)i * D * D, Wr + (long)i * D * D, b + (long)i * D,
          houts[i], N_NODES, relu, resid);
      h = houts[i];
    }
  }
}